// DifferentialEKVConv2d_12154757447733
// MI455X (gfx1250) — compile-verified
//
#include <hip/hip_runtime.h>
#include <stdint.h>

typedef float v2f __attribute__((ext_vector_type(2)));
typedef float v8f __attribute__((ext_vector_type(8)));

#define IN_CH 32
#define OUT_CH 64
#define HW 32
#define NB 4
#define KK 288           // IN_CH * K * K
#define XS_STRIDE 17     // bank-padded row stride for patch matrix
#define CHUNK 16         // output channels per block

// stable softplus: max(z,0) + log1p(exp(-|z|)), via native v_exp/v_log
__device__ __forceinline__ float softplus_f(float z) {
  float e = __expf(-fabsf(z));
  float l = __logf(1.0f + e);
  return fmaxf(z, 0.0f) + l;
}

// EKV current (without ALPHA scale): sp((x-th)/PHI)^2 - sp((x-th-VD)/PHI)^2
__device__ __forceinline__ float ekv(float x, float th) {
  const float INV_PHI = 12.820512820512821f;  // 1 / (2*1.5*0.026)
  const float VD_PHI  = 2.5641025641025643f;  // 0.2 / PHI
  float z  = (x - th) * INV_PHI;
  float zr = z - VD_PHI;
  float sf = softplus_f(z);
  float sr = softplus_f(zr);
  return fmaf(sf, sf, -(sr * sr));
}

// per-lane dynamic select out of the 8 D-matrix VGPRs (v_cndmask chain)
__device__ __forceinline__ float pick8(v8f d, int r) {
  float v = d[0];
  v = (r == 1) ? d[1] : v;
  v = (r == 2) ? d[2] : v;
  v = (r == 3) ? d[3] : v;
  v = (r == 4) ? d[4] : v;
  v = (r == 5) ? d[5] : v;
  v = (r == 6) ? d[6] : v;
  v = (r == 7) ? d[7] : v;
  return v;
}

// CDNA5 async global->LDS copy, 16 bytes per lane (tracked by ASYNCcnt)
__device__ __forceinline__ void async_ld16(unsigned lds_off, const void* g) {
  asm volatile("global_load_async_to_lds_b128 %0, %1, off"
               :: "v"(lds_off), "v"((unsigned long long)(uintptr_t)g)
               : "memory");
}

__global__ __launch_bounds__(256) void ekv_conv_kernel(
    const float* __restrict__ x, const float* __restrict__ theta_pos,
    const float* __restrict__ theta_neg, float* __restrict__ out) {
  __shared__ __align__(16) float xs[KK * XS_STRIDE];  // patch matrix [j][pixel]
  __shared__ __align__(16) float thp[CHUNK * KK];
  __shared__ __align__(16) float thn[CHUNK * KK];

  const int tid = threadIdx.x;
  const int bx = blockIdx.x;              // 0..255 : (n, h, w-half)
  const int cbase = blockIdx.y * CHUNK;   // channel chunk base

  const int n = bx >> 6;
  const int rem = bx & 63;
  const int h = rem >> 1;
  const int w0 = (rem & 1) * 16;

  // ---- stage theta chunk into LDS via CDNA5 async-to-LDS (contiguous) ----
  {
    const char* gp = (const char*)(theta_pos + cbase * KK);
    const char* gn = (const char*)(theta_neg + cbase * KK);
    unsigned lp = (unsigned)(uintptr_t)thp;
    unsigned ln = (unsigned)(uintptr_t)thn;
    for (int i = tid; i < (CHUNK * KK) / 4; i += 256) {  // 1152 x 16B each side
      async_ld16(lp + i * 16, gp + i * 16);
      async_ld16(ln + i * 16, gn + i * 16);
    }
  }

  // ---- build im2col patch matrix for this 16-pixel group (zero-padded) ----
  for (int e = tid; e < KK * 16; e += 256) {
    int j = e >> 4;
    int p = e & 15;
    int c = j / 9;
    int r9 = j - c * 9;
    int kh = r9 / 3;
    int kw = r9 - kh * 3;
    int hh = h + kh - 1;
    int ww = w0 + p + kw - 1;
    float v = 0.0f;
    if ((unsigned)hh < 32u && (unsigned)ww < 32u)
      v = x[((n * IN_CH + c) * HW + hh) * HW + ww];
    xs[j * XS_STRIDE + p] = v;
  }

  asm volatile("s_wait_asynccnt 0" ::: "memory");
  __syncthreads();

  const int lane = tid & 31;
  const int wv = tid >> 5;       // 0..7 -> 2 channels each
  const int half = lane >> 4;    // which 16-lane half
  const int m = lane & 15;       // pixel row (A-matrix M index)
  const int jb = half * 144;     // K-slots 2*half, 2*half+1 -> j bases

  #pragma unroll
  for (int s = 0; s < 2; ++s) {
    const int lc = wv * 2 + s;
    const float* tp = &thp[lc * KK];
    const float* tn = &thn[lc * KK];

    // 4 fp32 accumulators per pixel, laid out as WMMA f32 16x16x4 A-matrix:
    // lane m: a0=A[m,2*half], a1=A[m,2*half+1]; each slot owns 72 j's.
    float a0 = 0.0f, a1 = 0.0f;
    #pragma unroll 4
    for (int t = 0; t < 72; ++t) {
      int j0 = jb + t;
      int j1 = jb + 72 + t;
      float x0 = xs[j0 * XS_STRIDE + m];
      float x1 = xs[j1 * XS_STRIDE + m];
      a0 += ekv(x0, tp[j0]) - ekv(x0, tn[j0]);
      a1 += ekv(x1, tp[j1]) - ekv(x1, tn[j1]);
    }

    // exact fp32 reduction on the matrix pipe: D[m,n] = sum_k A[m,k] * 1
    v2f A; A[0] = a0; A[1] = a1;
    v2f B; B[0] = 1.0f; B[1] = 1.0f;
    v8f C = {};
    v8f D = __builtin_amdgcn_wmma_f32_16x16x4_f32(
        /*neg_a=*/false, A, /*neg_b=*/false, B,
        /*c_mod=*/(short)0, C, /*reuse_a=*/false, /*reuse_b=*/false);

    // D vgpr r: lanes 0-15 hold pixel r, lanes 16-31 hold pixel 8+r (all cols equal)
    int r = lane & 7;
    if ((lane & 15) < 8) {
      float val = pick8(D, r) * 0.02f;  // ALPHA * TIA_GAIN = 1e-5 * 2000
      int p = half * 8 + r;
      int o = cbase + lc;
      out[((n * OUT_CH + o) * HW + h) * HW + (w0 + p)] = val;
    }
  }
}

extern "C" void kernel_launch(void* const* d_in, const int* in_sizes, int n_in,
                              void* d_out, int out_size, void* d_ws, size_t ws_size,
                              hipStream_t stream) {
  const float* x  = (const float*)d_in[0];
  const float* tp = (const float*)d_in[1];
  const float* tn = (const float*)d_in[2];
  float* out = (float*)d_out;
  dim3 grid(NB * HW * 2, OUT_CH / CHUNK);  // (256, 4)
  ekv_conv_kernel<<<grid, 256, 0, stream>>>(x, tp, tn, out);
}